// GAEModel_58995670778279
// MI455X (gfx1250) — compile-verified
//
#include <hip/hip_runtime.h>
#include <hip/hip_bf16.h>
#include <math.h>

// ---------------- constants matching the reference ----------------
#define NNODES 50000
#define NEDGES 800000
#define NTOT   (NEDGES + NNODES)   // edges + self loops
#define F_IN   64
#define HDIM   256
#define EMB    128

typedef float v2f __attribute__((ext_vector_type(2)));
typedef float v8f __attribute__((ext_vector_type(8)));

__device__ __forceinline__ v8f wmma_f32(v2f a, v2f b, v8f c) {
    return __builtin_amdgcn_wmma_f32_16x16x4_f32(
        /*neg_a=*/false, a, /*neg_b=*/false, b,
        /*c_mod=*/(short)0, c, /*reuse_a=*/false, /*reuse_b=*/false);
}

// =================================================================
// GEMM: out[M x Ncol] = act(A[M x K] @ W[K x Ncol] (+ bias))
// M % 16 == 0, Ncol % 64 == 0, K % 4 == 0.
// One wave32 computes a 16x64 strip: 4 accumulators share one A-fragment
// per k-step (4x less A traffic, 4 independent V_WMMA_F32_16X16X4_F32
// per A-load -> no RAW chain between them).
// A-frag (32-bit A 16x4): lanes 0-15 / 16-31 hold M = lane%16;
//   vgpr0 = K={0|2}, vgpr1 = K={1|3} selected by lane-half.
// B-frag mirrors; C/D: vgpr i -> row = 8*half + i, col = lane%16.
// =================================================================
template <bool RELU, bool BIAS>
__global__ void gemm_wmma_f32(const float* __restrict__ A,
                              const float* __restrict__ W,
                              const float* __restrict__ bias,
                              float* __restrict__ out,
                              int M, int K, int Ncol) {
    const int wave   = (blockIdx.x * blockDim.x + threadIdx.x) >> 5;
    const int lane   = threadIdx.x & 31;
    const int tilesN = Ncol >> 6;                 // 64-col strips
    const int tileM  = wave / tilesN;
    const int tileN  = wave % tilesN;
    if (tileM * 16 >= M) return;                  // wave-uniform: EXEC stays all-1s

    const int m0   = tileM << 4;
    const int n0   = tileN << 6;
    const int r    = lane & 15;
    const int half = lane >> 4;

    const float* __restrict__ Arow = A + (size_t)(m0 + r) * K;
    const float* __restrict__ Wb   = W + (size_t)n0 + r;

    v8f acc0 = {}, acc1 = {}, acc2 = {}, acc3 = {};
#pragma unroll 4
    for (int k0 = 0; k0 < K; k0 += 4) {
        const int k = k0 + (half << 1);
        v2f a;
        a.x = Arow[k];
        a.y = Arow[k + 1];
        const float* __restrict__ w0 = Wb + (size_t)k * Ncol;
        const float* __restrict__ w1 = w0 + Ncol;
        v2f b0, b1, b2, b3;
        b0.x = w0[0];  b0.y = w1[0];
        b1.x = w0[16]; b1.y = w1[16];
        b2.x = w0[32]; b2.y = w1[32];
        b3.x = w0[48]; b3.y = w1[48];
        acc0 = wmma_f32(a, b0, acc0);
        acc1 = wmma_f32(a, b1, acc1);
        acc2 = wmma_f32(a, b2, acc2);
        acc3 = wmma_f32(a, b3, acc3);
    }

    float* __restrict__ orow = out + (size_t)(m0 + (half << 3)) * Ncol + n0 + r;
    const v8f accs[4] = {acc0, acc1, acc2, acc3};
#pragma unroll
    for (int t = 0; t < 4; ++t) {
        const float bv = BIAS ? bias[n0 + (t << 4) + r] : 0.0f;
#pragma unroll
        for (int i = 0; i < 8; ++i) {
            float v = accs[t][i] + bv;
            if (RELU) v = fmaxf(v, 0.0f);
            orow[(size_t)i * Ncol + (t << 4)] = v;
        }
    }
}

// ---------------- simple elementwise helpers ----------------
__global__ void zero_f32(float* __restrict__ p, int n) {
    int i = blockIdx.x * blockDim.x + threadIdx.x;
    if (i < n) p[i] = 0.0f;
}

// deg[c] += 1 for every target (edges then self-loops)
__global__ void deg_accum(const int* __restrict__ ei, float* __restrict__ deg) {
    int e = blockIdx.x * blockDim.x + threadIdx.x;
    if (e < NTOT) {
        int c = (e < NEDGES) ? ei[NEDGES + e] : (e - NEDGES);
        atomicAdd(&deg[c], 1.0f);
    }
}

__global__ void deg_to_dinv(float* __restrict__ deg) {
    int i = blockIdx.x * blockDim.x + threadIdx.x;
    if (i < NNODES) {
        float d = deg[i];
        deg[i] = (d > 0.0f) ? rsqrtf(d) : 0.0f;
    }
}

__global__ void norm_kernel(const int* __restrict__ ei,
                            const float* __restrict__ dinv,
                            float* __restrict__ nrm) {
    int e = blockIdx.x * blockDim.x + threadIdx.x;
    if (e < NTOT) {
        int r = (e < NEDGES) ? ei[e] : (e - NEDGES);
        int c = (e < NEDGES) ? ei[NEDGES + e] : (e - NEDGES);
        nrm[e] = dinv[r] * dinv[c];
    }
}

// out[n][f] = bias[f]  (H = 256 so f = i & 255)
__global__ void init_bias(float* __restrict__ o, const float* __restrict__ b) {
    size_t i = (size_t)blockIdx.x * blockDim.x + threadIdx.x;
    if (i < (size_t)NNODES * HDIM) o[i] = b[i & (HDIM - 1)];
}

__global__ void relu_inplace(float* __restrict__ o) {
    size_t i = (size_t)blockIdx.x * blockDim.x + threadIdx.x;
    if (i < (size_t)NNODES * HDIM) o[i] = fmaxf(o[i], 0.0f);
}

// scatter-add: o[col[e]][f] += t[row[e]][f] * norm[e]   (one block per edge)
__global__ void gcn_scatter(const float* __restrict__ t,
                            const int* __restrict__ ei,
                            const float* __restrict__ nrm,
                            float* __restrict__ o) {
    const int e = blockIdx.x;
    const int f = threadIdx.x;
    const int r = (e < NEDGES) ? ei[e] : (e - NEDGES);
    const int c = (e < NEDGES) ? ei[NEDGES + e] : (e - NEDGES);
    const float v = t[(size_t)r * HDIM + f] * nrm[e];
    atomicAdd(&o[(size_t)c * HDIM + f], v);
}

// ---------------- readout: mean + max over 50000 rows ----------------
#define POOL_BLOCKS 250
__global__ void pool_partial(const float* __restrict__ h,
                             float* __restrict__ psum, float* __restrict__ pmax) {
    const int f  = threadIdx.x;           // 256 features
    const int bb = blockIdx.x;            // 250 blocks, 200 rows each
    const int r0 = bb * (NNODES / POOL_BLOCKS);
    const int r1 = r0 + (NNODES / POOL_BLOCKS);
    float s = 0.0f, m = -INFINITY;
    for (int r = r0; r < r1; ++r) {
        float v = h[(size_t)r * HDIM + f];
        s += v;
        m = fmaxf(m, v);
    }
    psum[bb * HDIM + f] = s;
    pmax[bb * HDIM + f] = m;
}

__global__ void pool_final(const float* __restrict__ psum,
                           const float* __restrict__ pmax,
                           float* __restrict__ g) {
    const int f = threadIdx.x;
    float s = 0.0f, m = -INFINITY;
    for (int p = 0; p < POOL_BLOCKS; ++p) {
        s += psum[p * HDIM + f];
        m = fmaxf(m, pmax[p * HDIM + f]);
    }
    g[f]        = s * (1.0f / (float)NNODES);  // mean
    g[HDIM + f] = m;                            // max
}

// projection MLP: [1,512] -> relu(512x256) -> 256x128 ; tiny, one block
__global__ void proj_kernel(const float* __restrict__ g,
                            const float* __restrict__ W1, const float* __restrict__ b1,
                            const float* __restrict__ W2, const float* __restrict__ b2,
                            float* __restrict__ out) {
    __shared__ float s1[HDIM];
    const int j = threadIdx.x;            // 256 threads
    float acc = b1[j];
    for (int k = 0; k < 2 * HDIM; ++k) acc += g[k] * W1[(size_t)k * HDIM + j];
    s1[j] = fmaxf(acc, 0.0f);
    __syncthreads();
    if (j < EMB) {
        float o = b2[j];
        for (int k = 0; k < HDIM; ++k) o += s1[k] * W2[(size_t)k * EMB + j];
        out[j] = o;
    }
}

// edge decoder: one wave per edge, coalesced strided dot + wave32 xor-reduce
__global__ void edge_decode(const float* __restrict__ h,
                            const int* __restrict__ ei,
                            float* __restrict__ out) {
    const int wave = (blockIdx.x * blockDim.x + threadIdx.x) >> 5;
    const int lane = threadIdx.x & 31;
    if (wave >= NEDGES) return;
    const float* __restrict__ a = h + (size_t)ei[wave] * HDIM;
    const float* __restrict__ b = h + (size_t)ei[NEDGES + wave] * HDIM;
    float s = 0.0f;
#pragma unroll
    for (int f = lane; f < HDIM; f += 32) s += a[f] * b[f];
#pragma unroll
    for (int off = 16; off > 0; off >>= 1) s += __shfl_xor(s, off, 32);
    if (lane == 0) out[EMB + wave] = 1.0f / (1.0f + __expf(-s));
}

// =================================================================
extern "C" void kernel_launch(void* const* d_in, const int* in_sizes, int n_in,
                              void* d_out, int out_size, void* d_ws, size_t ws_size,
                              hipStream_t stream) {
    const float* x      = (const float*)d_in[0];
    const int*   ei     = (const int*)  d_in[1];
    const float* enc_W1 = (const float*)d_in[2];
    const float* enc_b1 = (const float*)d_in[3];
    const float* enc_W2 = (const float*)d_in[4];
    const float* enc_b2 = (const float*)d_in[5];
    const float* gcn_W0 = (const float*)d_in[6];
    const float* gcn_b0 = (const float*)d_in[7];
    const float* gcn_W1 = (const float*)d_in[8];
    const float* gcn_b1 = (const float*)d_in[9];
    const float* gcn_W2 = (const float*)d_in[10];
    const float* gcn_b2 = (const float*)d_in[11];
    const float* proj_W1 = (const float*)d_in[12];
    const float* proj_b1 = (const float*)d_in[13];
    const float* proj_W2 = (const float*)d_in[14];
    const float* proj_b2 = (const float*)d_in[15];
    float* out = (float*)d_out;

    // -------- workspace layout (floats); total ~39.4M floats (~158 MB) -----
    float* ws   = (float*)d_ws;
    const size_t NH = (size_t)NNODES * HDIM;          // 12.8M
    float* hA   = ws;
    float* hB   = hA + NH;
    float* hC   = hB + NH;
    float* deg  = hC + NH;                            // 50000 (also dinv)
    float* nrm  = deg + NNODES;                       // 850000
    float* psum = nrm + NTOT;                         // 250*256
    float* pmax = psum + POOL_BLOCKS * HDIM;
    float* g    = pmax + POOL_BLOCKS * HDIM;          // 512

    const int B = 256;
    const int gemm_tiles  = (NNODES / 16) * (HDIM / 64);     // 12500 strips
    const int gemm_blocks = (gemm_tiles + 7) / 8;            // 8 waves/block
    const int nh_blocks   = (int)((NH + B - 1) / B);

    // -------- encoder MLP --------
    // hA = relu(x @ enc_W1 + b1) ; hB = hA @ enc_W2 + b2
    gemm_wmma_f32<true,  true><<<gemm_blocks, B, 0, stream>>>(x,  enc_W1, enc_b1, hA, NNODES, F_IN, HDIM);
    gemm_wmma_f32<false, true><<<gemm_blocks, B, 0, stream>>>(hA, enc_W2, enc_b2, hB, NNODES, HDIM, HDIM);

    // -------- degree / symmetric norm (with self loops) --------
    zero_f32<<<(NNODES + B - 1) / B, B, 0, stream>>>(deg, NNODES);
    deg_accum<<<(NTOT + B - 1) / B, B, 0, stream>>>(ei, deg);
    deg_to_dinv<<<(NNODES + B - 1) / B, B, 0, stream>>>(deg);
    norm_kernel<<<(NTOT + B - 1) / B, B, 0, stream>>>(ei, deg, nrm);

    // -------- 3x GCN conv: t = h@W ; out = scatter(t*norm) + b ; relu ------
    // layer 0: hB -> hC
    gemm_wmma_f32<false, false><<<gemm_blocks, B, 0, stream>>>(hB, gcn_W0, nullptr, hA, NNODES, HDIM, HDIM);
    init_bias<<<nh_blocks, B, 0, stream>>>(hC, gcn_b0);
    gcn_scatter<<<NTOT, HDIM, 0, stream>>>(hA, ei, nrm, hC);
    relu_inplace<<<nh_blocks, B, 0, stream>>>(hC);
    // layer 1: hC -> hB
    gemm_wmma_f32<false, false><<<gemm_blocks, B, 0, stream>>>(hC, gcn_W1, nullptr, hA, NNODES, HDIM, HDIM);
    init_bias<<<nh_blocks, B, 0, stream>>>(hB, gcn_b1);
    gcn_scatter<<<NTOT, HDIM, 0, stream>>>(hA, ei, nrm, hB);
    relu_inplace<<<nh_blocks, B, 0, stream>>>(hB);
    // layer 2: hB -> hC   (final node embeddings in hC)
    gemm_wmma_f32<false, false><<<gemm_blocks, B, 0, stream>>>(hB, gcn_W2, nullptr, hA, NNODES, HDIM, HDIM);
    init_bias<<<nh_blocks, B, 0, stream>>>(hC, gcn_b2);
    gcn_scatter<<<NTOT, HDIM, 0, stream>>>(hA, ei, nrm, hC);
    relu_inplace<<<nh_blocks, B, 0, stream>>>(hC);

    // -------- readout + projection --------
    pool_partial<<<POOL_BLOCKS, HDIM, 0, stream>>>(hC, psum, pmax);
    pool_final<<<1, HDIM, 0, stream>>>(psum, pmax, g);
    proj_kernel<<<1, HDIM, 0, stream>>>(g, proj_W1, proj_b1, proj_W2, proj_b2, out);

    // -------- edge decoder --------
    const int dec_blocks = (NEDGES * 32 + B - 1) / B;   // one wave32 per edge
    edge_decode<<<dec_blocks, B, 0, stream>>>(hC, ei, out);
}